// DeformConv2d_9199819948337
// MI455X (gfx1250) — compile-verified
//
#include <hip/hip_runtime.h>

// Deformable conv2d (B=8, C=64, H=W=64, Cout=64, groups=4, 3x3, pad 1, stride 1)
// Pipeline per batch b (stream-ordered):
//   Kernel A: offset conv (grouped implicit GEMM, WMMA f32 16x16x4) -> off_ws[4096][1728]
//   Kernel B: group softmax + bilinear sample + grouped 16-chan GEMM (WMMA) -> out

typedef __attribute__((ext_vector_type(2))) float v2f;
typedef __attribute__((ext_vector_type(8))) float v8f;

#define H 64
#define W 64
#define C_IN 64
#define C_OUT 64
#define GROUPS 4
#define CG 16           // input channels per group
#define KTAPS 9
#define KRED 144        // CG * KTAPS : GEMM reduction depth
#define NOFF_G 432      // offset-conv out channels per group (= 9 tiles of 48)
#define NOFF 1728       // total offset-conv channels (3 * C * 9)
#define NPIX 4096       // H * W
#define AT_STRIDE 148   // LDS row stride (words). 148 % 64 == 20 -> the 16 low-half
                        // lanes hit the 16 distinct multiple-of-4 banks; high half
                        // (kb+2) hits the ==2 mod 4 banks. Conflict-free b64 reads.

__device__ __forceinline__ v8f wmma_f32(v2f a, v2f b, v8f c) {
  // D = A(16x4 f32) * B(4x16 f32) + C(16x16 f32)
  return __builtin_amdgcn_wmma_f32_16x16x4_f32(false, a, false, b, (short)0, c,
                                               false, false);
}

// ---------------------------------------------------------------------------
// Kernel A: grouped 3x3 offset conv, one batch.
// grid = (64 rows, 9 ntiles, 4 groups), block = 192 (6 waves).
// Block tile: M = 64 pixels (one output row) x N = 48 offset channels, K = 144.
// Wave w: ni = w%3 -> 16-chan column; mh = w/3 -> 32 pixels (two accumulators
// sharing one B fragment). No tail predicates anywhere.
// ---------------------------------------------------------------------------
__global__ __launch_bounds__(192)
void offconv_kernel(const float* __restrict__ inps,
                    const float* __restrict__ woff,
                    const float* __restrict__ boff,
                    float* __restrict__ off_ws, int b) {
  __shared__ float colT[W * AT_STRIDE];  // colT[pixel][k] : A-major layout
  const int oh = blockIdx.x;      // output row
  const int ntile = blockIdx.y;   // 48-channel tile within the group (0..8)
  const int g = blockIdx.z;       // conv group
  const int tid = threadIdx.x;
  const float* inb = inps + (size_t)b * C_IN * NPIX;

  // im2col tile: colT[p][j=c*9+tap] = in[g*16+c][oh+ky-1][p+kx-1], zero-padded.
  // p fastest -> coalesced global reads along x.
  for (int idx = tid; idx < KRED * W; idx += 192) {
    int j = idx >> 6, p = idx & 63;
    int c = j / 9, tap = j - c * 9;
    int ky = tap / 3, kx = tap - ky * 3;
    int y = oh + ky - 1, x = p + kx - 1;
    float v = 0.f;
    if (y >= 0 && y < H && x >= 0 && x < W)
      v = inb[((g * CG + c) * H + y) * W + x];
    colT[p * AT_STRIDE + j] = v;
  }
  __syncthreads();

  const int wav = tid >> 5, lane = tid & 31;
  const int ni = wav % 3, mh = wav / 3;   // chan column / pixel half
  const int lhalf = (lane & 16) ? 1 : 0;  // lanes 16-31 hold K+2,K+3
  const int l15 = lane & 15;
  const int nl = ntile * 48 + ni * 16 + l15;  // channel within group, < 432
  const int m0 = mh * 32 + l15;

  v8f c0 = {0.f, 0.f, 0.f, 0.f, 0.f, 0.f, 0.f, 0.f};
  v8f c1 = c0;
  const float* wrow = woff + (size_t)(g * NOFF_G + nl) * KRED + 2 * lhalf;
  const float* arow0 = colT + m0 * AT_STRIDE + 2 * lhalf;
  const float* arow1 = arow0 + 16 * AT_STRIDE;

#pragma unroll
  for (int kk = 0; kk < 36; ++kk) {
    v2f bf = *(const v2f*)(wrow + 4 * kk);   // global_load_b64, L2-resident
    v2f a0 = *(const v2f*)(arow0 + 4 * kk);  // ds_load_b64, conflict-free
    v2f a1 = *(const v2f*)(arow1 + 4 * kk);
    c0 = wmma_f32(a0, bf, c0);
    c1 = wmma_f32(a1, bf, c1);
  }

  const int chan = g * NOFF_G + nl;
  const float bo = boff[chan];
  const int mrow = mh * 32 + lhalf * 8;  // D: vgpr r -> M = r (+8 for hi lanes)
#pragma unroll
  for (int r = 0; r < 8; ++r) {
    int pix0 = oh * W + mrow + r;
    off_ws[(size_t)pix0 * NOFF + chan] = c0[r] + bo;  // coalesced in chan
    off_ws[(size_t)(pix0 + 16) * NOFF + chan] = c1[r] + bo;
  }
}

// ---------------------------------------------------------------------------
// Kernel B: group-softmax mask + bilinear sampling + grouped output GEMM.
// grid = (256 pixel tiles), block = 128 (4 waves, wave g handles group g).
// Per tile: S_g^T[16 pix][144] built in LDS, then D[16 pix x 16 chan] via
// 36 WMMA f32 16x16x4 steps.
// ---------------------------------------------------------------------------
__global__ __launch_bounds__(128)
void sample_gemm_kernel(const float* __restrict__ inps,
                        const float* __restrict__ weight,
                        const float* __restrict__ bias,
                        const float* __restrict__ off_ws,
                        float* __restrict__ out, int b) {
  __shared__ float smatT[GROUPS * 16 * AT_STRIDE];  // [g][pixel][k]
  const int p0 = blockIdx.x * 16;
  const int tid = threadIdx.x;
  const int g = tid >> 5, lane = tid & 31;
  const float* inb = inps + (size_t)b * C_IN * NPIX;

  // Phase 1: sampled & masked matrix for this wave's group.
  for (int idx = lane; idx < KRED * 16; idx += 32) {
    int j = idx >> 4, p = idx & 15;           // j = c_local*9 + tap
    int c = j / 9, tap = j - c * 9;
    int ky = tap / 3, kx = tap - ky * 3;
    int pix = p0 + p;
    int oh = pix >> 6, ow = pix & 63;
    const float* wsrow = off_ws + (size_t)pix * NOFF;
    // channel (g*16+c)*9+tap == g*144 + j : dy / dx / mask planes
    float dy = wsrow[g * KRED + j];
    float dx = wsrow[576 + g * KRED + j];
    float m0r = wsrow[1152 + 0 * KRED + j];
    float m1r = wsrow[1152 + 1 * KRED + j];
    float m2r = wsrow[1152 + 2 * KRED + j];
    float m3r = wsrow[1152 + 3 * KRED + j];
    float mx = fmaxf(fmaxf(m0r, m1r), fmaxf(m2r, m3r));
    float e0 = __expf(m0r - mx), e1 = __expf(m1r - mx);
    float e2 = __expf(m2r - mx), e3 = __expf(m3r - mx);
    float den = e0 + e1 + e2 + e3;
    float eg = (g == 0) ? e0 : (g == 1) ? e1 : (g == 2) ? e2 : e3;
    float mask = eg / den;

    // padded-coordinate sample position; padded ring is zero
    float py = (float)(oh + ky) + dy;
    float px = (float)(ow + kx) + dx;
    float y0 = floorf(py), x0 = floorf(px);
    float wy = py - y0, wx = px - x0;
    const float* chp = inb + (size_t)(g * CG + c) * NPIX;
    float acc = 0.f;
#pragma unroll
    for (int dyc = 0; dyc < 2; ++dyc) {
#pragma unroll
      for (int dxc = 0; dxc < 2; ++dxc) {
        float yy = y0 + (float)dyc, xx = x0 + (float)dxc;
        if (yy >= 1.f && yy <= 64.f && xx >= 1.f && xx <= 64.f) {
          float wgt = (dyc ? wy : 1.f - wy) * (dxc ? wx : 1.f - wx);
          acc += wgt * chp[((int)yy - 1) * W + ((int)xx - 1)];
        }
      }
    }
    smatT[(g * 16 + p) * AT_STRIDE + j] = acc * mask;
  }
  __syncthreads();

  // Phase 2: out_g[16 pix][16 chan] = S_g(16x144) * Wg^T(144x16)
  const int lhalf = (lane & 16) ? 1 : 0;
  const int l15 = lane & 15;
  const int chan = g * CG + l15;
  const float* wrow = weight + (size_t)chan * KRED + 2 * lhalf;
  const float* arow = smatT + (g * 16 + l15) * AT_STRIDE + 2 * lhalf;
  v8f acc = {0.f, 0.f, 0.f, 0.f, 0.f, 0.f, 0.f, 0.f};
#pragma unroll
  for (int kk = 0; kk < 36; ++kk) {
    v2f bf = *(const v2f*)(wrow + 4 * kk);  // B[k][n] = weight[n][k]
    v2f af = *(const v2f*)(arow + 4 * kk);  // A[m=pixel][k]
    acc = wmma_f32(af, bf, acc);
  }
  const float bs = bias[chan];
  const int prow = p0 + lhalf * 8;
#pragma unroll
  for (int r = 0; r < 8; ++r)
    out[((size_t)b * C_OUT + chan) * NPIX + prow + r] = acc[r] + bs;
}

extern "C" void kernel_launch(void* const* d_in, const int* in_sizes, int n_in,
                              void* d_out, int out_size, void* d_ws,
                              size_t ws_size, hipStream_t stream) {
  const float* inps = (const float*)d_in[0];        // (8,64,64,64)
  const float* weight = (const float*)d_in[1];      // (64,16,3,3)
  const float* bias = (const float*)d_in[2];        // (64,)
  const float* weight_off = (const float*)d_in[3];  // (1728,16,3,3)
  const float* bias_off = (const float*)d_in[4];    // (1728,)
  float* out = (float*)d_out;                       // (8,64,64,64)
  float* off_ws = (float*)d_ws;                     // 1728*4096 f32 = 28.3 MB

  for (int b = 0; b < 8; ++b) {
    offconv_kernel<<<dim3(64, 9, 4), 192, 0, stream>>>(inps, weight_off,
                                                       bias_off, off_ws, b);
    sample_gemm_kernel<<<dim3(256), 128, 0, stream>>>(inps, weight, bias,
                                                      off_ws, out, b);
  }
}